// GCN_26663156973940
// MI455X (gfx1250) — compile-verified
//
#include <hip/hip_runtime.h>

typedef __attribute__((ext_vector_type(2))) float v2f;
typedef __attribute__((ext_vector_type(8))) float v8f;

#define N_NODES 40000
#define N_EDGES 640000
#define DFEAT   128
#define NGRAPH  64
#define NCLS    10

// ---------------- degree / norm ----------------
__global__ void k_init_deg(float* __restrict__ deg) {
  int i = blockIdx.x * blockDim.x + threadIdx.x;
  if (i < N_NODES) deg[i] = 1.0f;            // self-loop contributes 1
}

__global__ void k_count_deg(const int* __restrict__ dst, float* __restrict__ deg) {
  int e = blockIdx.x * blockDim.x + threadIdx.x;
  if (e < N_EDGES) unsafeAtomicAdd(&deg[dst[e]], 1.0f);  // global_atomic_add_f32
}

__global__ void k_dinv(const float* __restrict__ deg, float* __restrict__ dinv) {
  int i = blockIdx.x * blockDim.x + threadIdx.x;
  if (i < N_NODES) dinv[i] = rsqrtf(deg[i]); // deg >= 1 always
}

// ---------------- WMMA f32 GEMM: out[Mx128] = act(A)[Mx128] @ W[128x128] ----------------
// 256 threads = 8 wave32; block b owns rows [16b,16b+16); wave w owns cols [16w,16w+16).
// FUSE_ACT: apply relu(v + bias) to A elements while staging into LDS.
template <bool FUSE_ACT>
__global__ __launch_bounds__(256) void k_gemm128_wmma(const float* __restrict__ A,
                                                      const float* __restrict__ bias,
                                                      const float* __restrict__ W,
                                                      float* __restrict__ out) {
  __shared__ float lX[16 * 132];   // 16 rows of A, K=0..127, stride 132 (bank-spread)
  __shared__ float lW[32 * 136];   // 32 K-rows of W, N=0..127, stride 136 (bank-spread)

  const int tid   = threadIdx.x;
  const int wave  = tid >> 5;
  const int lane  = tid & 31;
  const int row16 = lane & 15;
  const int hi    = lane >> 4;       // half-wave select (K+2 / M+8 per ISA layouts)
  const int m0    = blockIdx.x * 16;
  const int n0    = wave * 16;

  // stage 16x128 A strip (coalesced float4), optionally fusing bias+relu
  {
    const float4* a4 = (const float4*)(A + (size_t)m0 * DFEAT);
#pragma unroll
    for (int i = 0; i < 2; ++i) {
      int f   = tid + i * 256;       // float4 index 0..511
      int lin = f * 4;
      int r   = lin >> 7;
      int c   = lin & 127;
      float4 v = a4[f];
      if (FUSE_ACT) {
        float4 b = ((const float4*)bias)[c >> 2];
        v.x = fmaxf(v.x + b.x, 0.0f);
        v.y = fmaxf(v.y + b.y, 0.0f);
        v.z = fmaxf(v.z + b.z, 0.0f);
        v.w = fmaxf(v.w + b.w, 0.0f);
      }
      lX[r * 132 + c + 0] = v.x;
      lX[r * 132 + c + 1] = v.y;
      lX[r * 132 + c + 2] = v.z;
      lX[r * 132 + c + 3] = v.w;
    }
  }

  v8f acc = {};
  for (int kc = 0; kc < DFEAT; kc += 32) {
    __syncthreads();                 // A staged / previous lW consumed
    const float4* w4 = (const float4*)(W + (size_t)kc * DFEAT);
#pragma unroll
    for (int i = 0; i < 4; ++i) {
      int f   = tid + i * 256;       // float4 index 0..1023 (32 rows x 128 cols)
      int lin = f * 4;
      int r   = lin >> 7;
      int c   = lin & 127;
      float4 v = w4[f];
      lW[r * 136 + c + 0] = v.x;
      lW[r * 136 + c + 1] = v.y;
      lW[r * 136 + c + 2] = v.z;
      lW[r * 136 + c + 3] = v.w;
    }
    __syncthreads();

#pragma unroll
    for (int kk = 0; kk < 32; kk += 4) {
      // A 16x4: VGPR0 = (M=row16, K=base+2*hi), VGPR1 = K+1
      int ka = kc + kk + 2 * hi;
      v2f a, b;
      a.x = lX[row16 * 132 + ka];
      a.y = lX[row16 * 132 + ka + 1];
      // B 4x16: VGPR0 = (K=base+2*hi, N=n0+row16), VGPR1 = K+1
      b.x = lW[(kk + 2 * hi) * 136 + n0 + row16];
      b.y = lW[(kk + 2 * hi + 1) * 136 + n0 + row16];
      acc = __builtin_amdgcn_wmma_f32_16x16x4_f32(false, a, false, b,
                                                  (short)0, acc, false, false);
    }
  }

  // D 16x16 f32: VGPR r -> M = r + 8*hi, N = n0 + row16
#pragma unroll
  for (int r = 0; r < 8; ++r) {
    int m = m0 + r + 8 * hi;
    out[(size_t)m * DFEAT + n0 + row16] = acc[r];
  }
}

// ---------------- aggregation: out[d] += norm * xw[s] (+ self loop) ----------------
__global__ void k_agg_self(const float* __restrict__ xw, const float* __restrict__ dinv,
                           float* __restrict__ out) {
  int t = blockIdx.x * blockDim.x + threadIdx.x;   // N_NODES*32 float4 lanes
  if (t >= N_NODES * 32) return;
  int node = t >> 5;
  float nn = dinv[node]; nn *= nn;                 // norm of self-loop = dinv^2
  float4 v = ((const float4*)xw)[t];
  v.x *= nn; v.y *= nn; v.z *= nn; v.w *= nn;
  ((float4*)out)[t] = v;                           // initializes out
}

__global__ void k_agg_edges(const float* __restrict__ xw, const int* __restrict__ src,
                            const int* __restrict__ dst, const float* __restrict__ dinv,
                            float* __restrict__ out) {
  int t = blockIdx.x * blockDim.x + threadIdx.x;   // N_EDGES*32 float4 lanes
  if (t >= N_EDGES * 32) return;
  int e  = t >> 5;
  int c4 = t & 31;
  int s = src[e], d = dst[e];
  float norm = dinv[s] * dinv[d];
  float4 v = ((const float4*)xw)[s * 32 + c4];     // gather (L2-resident)
  float* o = out + (size_t)d * DFEAT + c4 * 4;
  unsafeAtomicAdd(o + 0, v.x * norm);
  unsafeAtomicAdd(o + 1, v.y * norm);
  unsafeAtomicAdd(o + 2, v.z * norm);
  unsafeAtomicAdd(o + 3, v.w * norm);
}

__global__ void k_bias_relu(float* __restrict__ h, const float* __restrict__ bias) {
  int t = blockIdx.x * blockDim.x + threadIdx.x;   // N_NODES*32
  if (t >= N_NODES * 32) return;
  int c4 = t & 31;
  float4 b = ((const float4*)bias)[c4];
  float4 v = ((float4*)h)[t];
  v.x = fmaxf(v.x + b.x, 0.0f);
  v.y = fmaxf(v.y + b.y, 0.0f);
  v.z = fmaxf(v.z + b.z, 0.0f);
  v.w = fmaxf(v.w + b.w, 0.0f);
  ((float4*)h)[t] = v;
}

// ---------------- pooling + head ----------------
__global__ void k_pool_zero(float* __restrict__ pool, float* __restrict__ cnt) {
  int t = blockIdx.x * blockDim.x + threadIdx.x;
  if (t < NGRAPH * DFEAT) pool[t] = 0.0f;
  if (t < NGRAPH) cnt[t] = 0.0f;
}

__global__ void k_pool_accum(const float* __restrict__ h, const int* __restrict__ batch,
                             float* __restrict__ pool, float* __restrict__ cnt) {
  int t = blockIdx.x * blockDim.x + threadIdx.x;   // N_NODES*32
  if (t >= N_NODES * 32) return;
  int node = t >> 5;
  int c4   = t & 31;
  int b = batch[node];
  float4 v = ((const float4*)h)[t];
  float* p = pool + (size_t)b * DFEAT + c4 * 4;
  unsafeAtomicAdd(p + 0, v.x);
  unsafeAtomicAdd(p + 1, v.y);
  unsafeAtomicAdd(p + 2, v.z);
  unsafeAtomicAdd(p + 3, v.w);
  if (c4 == 0) unsafeAtomicAdd(&cnt[b], 1.0f);
}

__global__ void k_logits(const float* __restrict__ pool, const float* __restrict__ cnt,
                         const float* __restrict__ lw, const float* __restrict__ lb,
                         float* __restrict__ out) {
  int t = blockIdx.x * blockDim.x + threadIdx.x;   // 64*10
  if (t >= NGRAPH * NCLS) return;
  int b = t / NCLS, k = t % NCLS;
  float c = fmaxf(cnt[b], 1.0f);
  float sum = 0.0f;
  for (int j = 0; j < DFEAT; ++j)
    sum += pool[b * DFEAT + j] * lw[j * NCLS + k];
  out[t] = sum / c + lb[k];
}

// ---------------- driver ----------------
extern "C" void kernel_launch(void* const* d_in, const int* in_sizes, int n_in,
                              void* d_out, int out_size, void* d_ws, size_t ws_size,
                              hipStream_t stream) {
  const float* x     = (const float*)d_in[0];
  const int*   ei    = (const int*)  d_in[1];
  const int*   batch = (const int*)  d_in[2];
  const float* W1    = (const float*)d_in[3];
  const float* b1    = (const float*)d_in[4];
  const float* W2    = (const float*)d_in[5];
  const float* b2    = (const float*)d_in[6];
  const float* lw    = (const float*)d_in[7];
  const float* lb    = (const float*)d_in[8];
  float* out = (float*)d_out;

  char* ws = (char*)d_ws;
  float* bufA = (float*)(ws);                  // 40000*128 f32 = 20,480,000 B
  float* bufB = (float*)(ws + 20480000);       // 20,480,000 B
  float* deg  = (float*)(ws + 40960000);       // 160,000 B
  float* dinv = (float*)(ws + 41120000);       // 160,000 B
  float* pool = (float*)(ws + 41280000);       // 32,768 B
  float* cnt  = (float*)(ws + 41312768);       // 256 B

  const int* src = ei;                         // edge_index[0]
  const int* dst = ei + N_EDGES;               // edge_index[1]

  const int B = 256;
  const int gN  = (N_NODES + B - 1) / B;
  const int gE  = (N_EDGES + B - 1) / B;
  const int gNL = (N_NODES * 32 + B - 1) / B;  // float4 lanes over node features
  const int gEL = (N_EDGES * 32 + B - 1) / B;  // float4 lanes over edge messages

  // normalization
  k_init_deg <<<gN, B, 0, stream>>>(deg);
  k_count_deg<<<gE, B, 0, stream>>>(dst, deg);
  k_dinv     <<<gN, B, 0, stream>>>(deg, dinv);

  // layer 1: z1 = D^-1/2 (A+I) D^-1/2 (x @ W1)   (bias+relu deferred into layer-2 GEMM)
  k_gemm128_wmma<false><<<N_NODES / 16, B, 0, stream>>>(x, nullptr, W1, bufA);
  k_agg_self    <<<gNL, B, 0, stream>>>(bufA, dinv, bufB);
  k_agg_edges   <<<gEL, B, 0, stream>>>(bufA, src, dst, dinv, bufB);

  // layer 2: h2 = relu(D^-1/2 (A+I) D^-1/2 (relu(z1 + b1) @ W2) + b2)
  k_gemm128_wmma<true><<<N_NODES / 16, B, 0, stream>>>(bufB, b1, W2, bufA);
  k_agg_self    <<<gNL, B, 0, stream>>>(bufA, dinv, bufB);
  k_agg_edges   <<<gEL, B, 0, stream>>>(bufA, src, dst, dinv, bufB);
  k_bias_relu   <<<gNL, B, 0, stream>>>(bufB, b2);

  // mean pool per graph + linear head
  k_pool_zero <<<(NGRAPH * DFEAT + B - 1) / B, B, 0, stream>>>(pool, cnt);
  k_pool_accum<<<gNL, B, 0, stream>>>(bufB, batch, pool, cnt);
  k_logits    <<<(NGRAPH * NCLS + B - 1) / B, B, 0, stream>>>(pool, cnt, lw, lb, out);
}